// SoftSeparationIPMModule_6262062317836
// MI455X (gfx1250) — compile-verified
//
#include <hip/hip_runtime.h>
#include <math.h>

// Problem constants from the reference
#define Bn    16
#define Nn    12
#define Mm    11          // Nn - 1 constraints
#define Wn    2048
#define BETA1 0.5f
#define BETA2 0.055f
#define EPSI  0.01f
#define MAXIT 50
#define TPB   256         // 8 wave32 per workgroup; 8 blocks per batch row

typedef float v2f __attribute__((ext_vector_type(2)));
typedef float v8f __attribute__((ext_vector_type(8)));

// ---------------------------------------------------------------------------
// Hash-based uniforms: jax.random.uniform(key 42/43) is not bit-reproducible
// outside JAX; we draw from the same distribution deterministically.
// ---------------------------------------------------------------------------
__device__ __forceinline__ unsigned hash_u32(unsigned x) {
  x ^= x >> 16; x *= 0x7feb352du;
  x ^= x >> 15; x *= 0x846ca68bu;
  x ^= x >> 16; return x;
}
__device__ __forceinline__ float unif01(unsigned seed) {
  // strictly inside (0,1)
  return ((float)(hash_u32(seed) >> 8) + 0.5f) * (1.0f / 16777216.0f);
}

// Residual norm || [R1; R2] ||_2 at (S + a*dS, Lam + a*dL) with fixed t.
// H is tridiagonal (hd/ho), (A^T L)_n = L_n - L_{n-1}, (A S)_i = S_i - S_{i+1}.
__device__ __forceinline__ float res_norm(
    const float* hd, const float* ho, const float* hb,
    const float* S,  const float* L,
    const float* dS, const float* dL, float a, float invt)
{
  float Sx[Nn], Lx[Mm];
#pragma unroll
  for (int n = 0; n < Nn; ++n) Sx[n] = S[n] + a * dS[n];
#pragma unroll
  for (int i = 0; i < Mm; ++i) Lx[i] = L[i] + a * dL[i];
  float acc = 0.f;
#pragma unroll
  for (int n = 0; n < Nn; ++n) {
    float h = hd[n] * Sx[n] + hb[n];
    if (n > 0)  h += ho[n - 1] * Sx[n - 1];
    if (n < Mm) h += ho[n] * Sx[n + 1];
    if (n < Mm) h += Lx[n];
    if (n > 0)  h -= Lx[n - 1];
    acc += h * h;
  }
#pragma unroll
  for (int i = 0; i < Mm; ++i) {
    float asd = Sx[i] - Sx[i + 1];
    float r = -Lx[i] * asd - invt;
    acc += r * r;
  }
  return sqrtf(acc);
}

// A (padded to 16x16): A[i][j] = [j==i] - [j==i+1] for i<11, zero rows above.
__device__ __forceinline__ float adiff(int i, int j) {
  if (i >= Mm) return 0.f;
  return (j == i) ? 1.f : ((j == i + 1) ? -1.f : 0.f);
}

__global__ __launch_bounds__(TPB)
void ipm_kernel(const float* __restrict__ Mu, const float* __restrict__ Sg,
                const float* __restrict__ Rr, float* __restrict__ out,
                float* __restrict__ ws)
{
  // 35 rows x 256 lanes of f32 staged via async DMA (35 KB of the 320 KB WGP LDS)
  __shared__ float shm[(2 * Nn + Mm) * TPB];

  const int tid = threadIdx.x;
  const int bb  = blockIdx.x >> 3;          // batch row (uniform -> SGPR base)
  const int w0  = (blockIdx.x & 7) * TPB;   // w-tile origin (uniform)

  // ---- async stage: Mu[12 rows], sigma2[12 rows], R[11 rows] -> LDS --------
  {
    unsigned voff = (unsigned)tid * 4u;
    const float* mrow = Mu + ((size_t)bb * Nn) * Wn + w0;
#pragma unroll
    for (int n = 0; n < Nn; ++n) {
      unsigned lo = (unsigned)(size_t)(&shm[n * TPB + tid]);
      asm volatile("global_load_async_to_lds_b32 %0, %1, %2"
                   :: "v"(lo), "v"(voff), "s"(mrow + (size_t)n * Wn) : "memory");
    }
    const float* srow = Sg + ((size_t)bb * Nn) * Wn + w0;
#pragma unroll
    for (int n = 0; n < Nn; ++n) {
      unsigned lo = (unsigned)(size_t)(&shm[(Nn + n) * TPB + tid]);
      asm volatile("global_load_async_to_lds_b32 %0, %1, %2"
                   :: "v"(lo), "v"(voff), "s"(srow + (size_t)n * Wn) : "memory");
    }
    const float* rrow = Rr + ((size_t)bb * Mm) * Wn + w0;
#pragma unroll
    for (int i = 0; i < Mm; ++i) {
      unsigned lo = (unsigned)(size_t)(&shm[(2 * Nn + i) * TPB + tid]);
      asm volatile("global_load_async_to_lds_b32 %0, %1, %2"
                   :: "v"(lo), "v"(voff), "s"(rrow + (size_t)i * Wn) : "memory");
    }
    asm volatile("s_wait_asynccnt 0" ::: "memory");
  }
  __syncthreads();

  // ---- per-lane QP build (one problem per lane) ----------------------------
  float muv[Nn], sgv[Nn], rv[Mm];
#pragma unroll
  for (int n = 0; n < Nn; ++n) muv[n] = shm[n * TPB + tid];
#pragma unroll
  for (int n = 0; n < Nn; ++n) sgv[n] = shm[(Nn + n) * TPB + tid];
#pragma unroll
  for (int i = 0; i < Mm; ++i) rv[i] = shm[(2 * Nn + i) * TPB + tid];

  float cmax = sgv[0];
#pragma unroll
  for (int n = 1; n < Nn; ++n) cmax = fmaxf(cmax, sgv[n]);
  cmax *= 20.0f;
#pragma unroll
  for (int n = 0; n < Nn; ++n) sgv[n] += 1e-8f;

  float pc[Mm];
#pragma unroll
  for (int i = 0; i < Mm; ++i) pc[i] = sgv[i + 1] / (sgv[i + 1] + sgv[i]);

  float hd[Nn], ho[Mm], hb[Nn];
#pragma unroll
  for (int n = 0; n < Nn; ++n) hd[n] = cmax / sgv[n];
#pragma unroll
  for (int i = 0; i < Mm; ++i) {
    hd[i]     += 2.0f * pc[i];
    hd[i + 1] += 2.0f * pc[i];
    ho[i]      = -2.0f * pc[i];
  }
#pragma unroll
  for (int n = 0; n < Nn; ++n) hb[n] = -cmax * muv[n] / sgv[n];
#pragma unroll
  for (int i = 0; i < Mm; ++i) {
    hb[i + 1] += -2.0f * pc[i] * rv[i];
    hb[i]     +=  2.0f * pc[i] * rv[i];
  }

  // S0 = running max of Mu along N (associative_scan maximum)
  float S[Nn];
  S[0] = muv[0];
#pragma unroll
  for (int n = 1; n < Nn; ++n) S[n] = fmaxf(S[n - 1], muv[n]);

  const unsigned p = blockIdx.x * TPB + tid;  // global problem id
  float Lam[Mm];
#pragma unroll
  for (int i = 0; i < Mm; ++i) Lam[i] = unif01(p * 32u + (unsigned)i + 1u);
  const float beta3 = 10.0f + unif01(p * 32u + 17u);

  // ---- IPM loop. Schur complement of the banded KKT system is tridiagonal:
  //   (H - A^T diag(Lam_i/ASd_i) A) dS = -R1 - A^T (R2/ASd)
  // solved by the Thomas algorithm; dLam recovered by back-substitution.
  // (Per-problem convergence; the reference gates on the *global* max residual
  //  with the same 51-iteration cap.)
  float r2 = 3.4e38f;
  for (int it = 0; it <= MAXIT && r2 >= EPSI; ++it) {
    float ASd[Mm];
#pragma unroll
    for (int i = 0; i < Mm; ++i) ASd[i] = S[i] - S[i + 1];   // d == 0

    float dot = 0.f;
#pragma unroll
    for (int i = 0; i < Mm; ++i) dot += ASd[i] * Lam[i];
    const float t    = -beta3 * (float)Mm / dot;
    const float invt = 1.0f / t;

    float R1[Nn], R2v[Mm];
#pragma unroll
    for (int n = 0; n < Nn; ++n) {
      float h = hd[n] * S[n] + hb[n];
      if (n > 0)  h += ho[n - 1] * S[n - 1];
      if (n < Mm) h += ho[n] * S[n + 1];
      if (n < Mm) h += Lam[n];
      if (n > 0)  h -= Lam[n - 1];
      R1[n] = h;
    }
#pragma unroll
    for (int i = 0; i < Mm; ++i) R2v[i] = -Lam[i] * ASd[i] - invt;

    float g[Mm], q[Mm];
#pragma unroll
    for (int i = 0; i < Mm; ++i) { g[i] = Lam[i] / ASd[i]; q[i] = R2v[i] / ASd[i]; }

    float Td[Nn], Tl[Mm], rh[Nn];
#pragma unroll
    for (int n = 0; n < Nn; ++n) {
      float gs = 0.f, qs = 0.f;
      if (n < Mm) { gs += g[n];     qs += q[n]; }
      if (n > 0)  { gs += g[n - 1]; qs -= q[n - 1]; }
      Td[n] = hd[n] - gs;
      rh[n] = -R1[n] - qs;
    }
#pragma unroll
    for (int i = 0; i < Mm; ++i) Tl[i] = ho[i] + g[i];

    // Thomas solve (no pivoting; matches the reference's unpivoted algebra)
    float cp[Mm], dp[Nn];
    cp[0] = Tl[0] / Td[0];
    dp[0] = rh[0] / Td[0];
#pragma unroll
    for (int n = 1; n < Nn; ++n) {
      float inv = 1.0f / (Td[n] - Tl[n - 1] * cp[n - 1]);
      if (n < Mm) cp[n] = Tl[n] * inv;
      dp[n] = (rh[n] - Tl[n - 1] * dp[n - 1]) * inv;
    }
    float dS[Nn];
    dS[Nn - 1] = dp[Nn - 1];
#pragma unroll
    for (int n = Nn - 2; n >= 0; --n) dS[n] = dp[n] - cp[n] * dS[n + 1];

    float dL[Mm];
#pragma unroll
    for (int i = 0; i < Mm; ++i)
      dL[i] = (R2v[i] - Lam[i] * (dS[i] - dS[i + 1])) / ASd[i];

    // fraction-to-boundary step
    float amin = 3.4e38f;
#pragma unroll
    for (int i = 0; i < Mm; ++i) {
      float np_ = (dL[i] < 0.f) ? dL[i] : -Lam[i];
      float a = -(Lam[i] / np_);
      if (!(a == a)) a = 1.0f;            // NaN -> 1 per reference
      amin = fminf(amin, a);
    }
    float alpha = 0.99f * amin;

    // feasibility backtracking (bounded; reference loop is unbounded)
    for (int k = 0; k < 64; ++k) {
      bool viol = false;
#pragma unroll
      for (int i = 0; i < Mm; ++i)
        viol |= ((S[i] + alpha * dS[i]) - (S[i + 1] + alpha * dS[i + 1]) > 0.f);
      if (!viol) break;
      alpha *= BETA1;
    }

    float rn0 = 0.f;
#pragma unroll
    for (int n = 0; n < Nn; ++n) rn0 += R1[n] * R1[n];
#pragma unroll
    for (int i = 0; i < Mm; ++i) rn0 += R2v[i] * R2v[i];
    rn0 = sqrtf(rn0);

    // Armijo-style residual backtracking
    float rn;
    for (int k = 0;; ++k) {
      rn = res_norm(hd, ho, hb, S, Lam, dS, dL, alpha, invt);
      if (rn <= (1.0f - BETA2 * alpha) * rn0 || k >= 63) break;
      alpha *= BETA1;
    }

#pragma unroll
    for (int n = 0; n < Nn; ++n) S[n] += alpha * dS[n];
#pragma unroll
    for (int i = 0; i < Mm; ++i) Lam[i] += alpha * dL[i];
    r2 = rn;
  }

  // ---- write output: out[(bb*N + n)*W + w] = S[n] --------------------------
#pragma unroll
  for (int n = 0; n < Nn; ++n)
    out[((size_t)bb * Nn + n) * Wn + w0 + tid] = S[n];

  // ---- WMMA diagnostic: batched feasibility gap max(A·S) -------------------
  // A is shared across all problems, so 16 problems' S-vectors form a dense
  // 16x16 B operand; chained V_WMMA_F32_16X16X4_F32 gives A·S for 16 columns.
  // Lane/VGPR layout per ISA 7.12.2 (A 16x4: V0={K0|K2}, V1={K1|K3};
  // B mirrored; C/D rows M and M+8 split across lane halves).
#pragma unroll
  for (int n = 0; n < Nn; ++n) shm[n * TPB + tid] = S[n];
  __syncthreads();

  const int lane = tid & 31, wid = tid >> 5;
  const int m16 = lane & 15, hi = lane >> 4;
  float gap = -3.4e38f;
#pragma unroll
  for (int pass = 0; pass < 2; ++pass) {
    v8f c = {};
#pragma unroll
    for (int k0 = 0; k0 < 16; k0 += 4) {
      v2f av, bv;
#pragma unroll
      for (int v = 0; v < 2; ++v) {
        int k = k0 + v + 2 * hi;
        av[v] = adiff(m16, k);
        int col = wid * 32 + pass * 16 + m16;           // problem column in LDS
        bv[v] = (k < Nn) ? shm[k * TPB + col] : 0.f;
      }
      c = __builtin_amdgcn_wmma_f32_16x16x4_f32(
              /*neg_a=*/false, av, /*neg_b=*/false, bv,
              /*c_mod=*/(short)0, c, /*reuse_a=*/false, /*reuse_b=*/false);
    }
#pragma unroll
    for (int e = 0; e < 8; ++e) gap = fmaxf(gap, c[e]);
  }
#pragma unroll
  for (int msk = 16; msk >= 1; msk >>= 1)
    gap = fmaxf(gap, __shfl_xor(gap, msk, 32));
  if (ws != nullptr && lane == 0)
    ws[(size_t)blockIdx.x * 8 + wid] = gap;   // 1024 floats of diagnostics
}

extern "C" void kernel_launch(void* const* d_in, const int* in_sizes, int n_in,
                              void* d_out, int out_size, void* d_ws, size_t ws_size,
                              hipStream_t stream) {
  (void)in_sizes; (void)n_in; (void)out_size;
  const float* Mu = (const float*)d_in[0];
  const float* Sg = (const float*)d_in[1];
  const float* Rr = (const float*)d_in[2];
  float* out = (float*)d_out;
  float* ws  = (ws_size >= (size_t)Bn * (Wn / TPB) * 8 * sizeof(float))
                   ? (float*)d_ws : nullptr;
  dim3 grid(Bn * (Wn / TPB));   // 128 workgroups
  dim3 block(TPB);              // 8 wave32 each -> 1024 waves total
  ipm_kernel<<<grid, block, 0, stream>>>(Mu, Sg, Rr, out, ws);
}